// InteractionLayer_35416300323238
// MI455X (gfx1250) — compile-verified
//
#include <hip/hip_runtime.h>

// CDNA5 / gfx1250: wave32, WMMA 16x16x4 f32.
typedef __attribute__((ext_vector_type(2))) float v2f;
typedef __attribute__((ext_vector_type(8))) float v8f;

#define NFEAT 27        // F = 26 sparse + 1
#define EDIM 128        // D
#define NPAIR 351       // F*(F-1)/2, strict upper triangle
#define ROWS_PAD 32
#define ROW_STRIDE 132  // floats: 128 + 4 pad -> 64-bank conflict-free b64 reads
#define SAMPLE_STRIDE (ROWS_PAD * ROW_STRIDE)   // 4224 floats
#define WAVES_PER_BLOCK 2
#define BLOCK_THREADS (WAVES_PER_BLOCK * 32)

__global__ __launch_bounds__(BLOCK_THREADS)
void interaction_triu_kernel(const float* __restrict__ emb,
                             float* __restrict__ out,
                             int nbatch)
{
    __shared__ float lds[WAVES_PER_BLOCK * SAMPLE_STRIDE]; // 33,792 bytes

    const int tid  = threadIdx.x;
    const int wave = tid >> 5;
    const int lane = tid & 31;
    const long long blockSample = (long long)blockIdx.x * WAVES_PER_BLOCK;

    // ---- Stage 2 samples into LDS (fully coalesced float4 global loads) ----
    const int NV4 = WAVES_PER_BLOCK * NFEAT * (EDIM / 4); // 1728
    for (int i = tid; i < NV4; i += BLOCK_THREADS) {
        const int s   = i / (NFEAT * (EDIM / 4));
        const int rem = i - s * (NFEAT * (EDIM / 4));
        const int row = rem >> 5;   // / (EDIM/4)
        const int c4  = rem & 31;
        const long long samp = blockSample + s;
        float4 v = make_float4(0.f, 0.f, 0.f, 0.f);
        if (samp < nbatch) {
            v = ((const float4*)emb)[samp * (NFEAT * (EDIM / 4)) + row * (EDIM / 4) + c4];
        }
        float* dst = &lds[s * SAMPLE_STRIDE + row * ROW_STRIDE + c4 * 4];
        dst[0] = v.x; dst[1] = v.y; dst[2] = v.z; dst[3] = v.w;
    }
    // Zero the pad rows 27..31 (WMMA then accumulates exact zeros there)
    const int NZ4 = WAVES_PER_BLOCK * (ROWS_PAD - NFEAT) * (EDIM / 4); // 320
    for (int i = tid; i < NZ4; i += BLOCK_THREADS) {
        const int s   = i / ((ROWS_PAD - NFEAT) * (EDIM / 4));
        const int rem = i - s * ((ROWS_PAD - NFEAT) * (EDIM / 4));
        const int row = NFEAT + (rem >> 5);
        const int c4  = rem & 31;
        float* dst = &lds[s * SAMPLE_STRIDE + row * ROW_STRIDE + c4 * 4];
        dst[0] = 0.f; dst[1] = 0.f; dst[2] = 0.f; dst[3] = 0.f;
    }
    __syncthreads();

    // ---- Compute: one wave owns one sample; Gram = E * E^T via f32 WMMA ----
    const long long samp = blockSample + wave;
    if (samp >= nbatch) return;   // whole-wave uniform branch; EXEC stays all-1 for WMMA

    const float* sm = &lds[wave * SAMPLE_STRIDE];
    const int r    = lane & 15;
    const int half = lane >> 4;   // 0 -> K pair {0,1}; 1 -> K pair {2,3}

    v8f c00 = {}; v8f c01 = {}; v8f c11 = {};
#pragma unroll
    for (int kk = 0; kk < EDIM / 4; ++kk) {
        const int col = 4 * kk + 2 * half;
        // Symmetric problem: the same fragment layout serves as A (rows) and B (cols^T).
        const v2f a0 = *(const v2f*)&sm[(r     ) * ROW_STRIDE + col];
        const v2f a1 = *(const v2f*)&sm[(r + 16) * ROW_STRIDE + col];
        c00 = __builtin_amdgcn_wmma_f32_16x16x4_f32(false, a0, false, a0, (short)0, c00, false, false);
        c01 = __builtin_amdgcn_wmma_f32_16x16x4_f32(false, a0, false, a1, (short)0, c01, false, false);
        c11 = __builtin_amdgcn_wmma_f32_16x16x4_f32(false, a1, false, a1, (short)0, c11, false, false);
    }

    // ---- Writeout: strict upper triangle, row-major triu order ----
    // C/D layout: lane = N + 16*half, element v of the v8f is M = v + 8*half.
    float* o = out + samp * NPAIR;
    const int n = lane & 15;
#pragma unroll
    for (int v = 0; v < 8; ++v) {
        // tile (0,0)
        {
            const int i = v + 8 * half;
            const int j = n;
            if (i < j) { // both < 16 < NFEAT
                o[i * (2 * NFEAT - i - 1) / 2 + (j - i - 1)] = c00[v];
            }
        }
        // tile (0,1)
        {
            const int i = v + 8 * half;
            const int j = 16 + n;
            if (j < NFEAT) { // i < 16 <= j always
                o[i * (2 * NFEAT - i - 1) / 2 + (j - i - 1)] = c01[v];
            }
        }
        // tile (1,1)
        {
            const int i = 16 + v + 8 * half;
            const int j = 16 + n;
            if (i < NFEAT && j < NFEAT && i < j) {
                o[i * (2 * NFEAT - i - 1) / 2 + (j - i - 1)] = c11[v];
            }
        }
    }
}

extern "C" void kernel_launch(void* const* d_in, const int* in_sizes, int n_in,
                              void* d_out, int out_size, void* d_ws, size_t ws_size,
                              hipStream_t stream) {
    const float* emb = (const float*)d_in[0];
    float* out = (float*)d_out;
    const int nbatch = in_sizes[0] / (NFEAT * EDIM); // 16384
    const int nblocks = (nbatch + WAVES_PER_BLOCK - 1) / WAVES_PER_BLOCK;
    hipLaunchKernelGGL(interaction_triu_kernel,
                       dim3(nblocks), dim3(BLOCK_THREADS), 0, stream,
                       emb, out, nbatch);
}